// MHGCN_douban_10187662426197
// MI455X (gfx1250) — compile-verified
//
#include <hip/hip_runtime.h>

#define N 8192
#define NFEAT 256
#define OUT 128
#define TOK 10

typedef __attribute__((ext_vector_type(2))) float v2f;
typedef __attribute__((ext_vector_type(8))) float v8f;

// D = A(16x4,f32) * B(4x16,f32) + C  -> v_wmma_f32_16x16x4_f32
__device__ __forceinline__ v8f wmma4(v2f a, v2f b, v8f c) {
  return __builtin_amdgcn_wmma_f32_16x16x4_f32(false, a, false, b, (short)0, c,
                                               false, false);
}

// A fragment (16x4): lane L holds row m=L%16, K pair kh=(L/16)*2 .. kh+1
// (row-major LDS tile, contiguous K pair -> single ds_load_b64)
__device__ __forceinline__ v2f ldsA(const float* s, int lda, int row0, int k0,
                                    int lane) {
  int m = lane & 15;
  int kh = (lane >> 4) << 1;
  const float* p = s + (row0 + m) * lda + k0 + kh;
  v2f a;
  a.x = p[0];
  a.y = p[1];
  return a;
}

// B fragment (4x16) from K-pair-packed LDS:  Bp[kpair*256 + n*2 + parity]
// lane L needs B[k][n], B[k+1][n] with k = k0 + (L/16)*2 -> one ds_load_b64
__device__ __forceinline__ v2f ldsBP(const float* s, int k0, int col0,
                                     int lane) {
  int n = lane & 15;
  int kp = (k0 >> 1) + (lane >> 4);
  const float* p = s + kp * 256 + ((col0 + n) << 1);
  v2f b;
  b.x = p[0];
  b.y = p[1];
  return b;
}

// ---------------------------------------------------------------------------
// C[N x OUT] = A[N x K] @ B[K x OUT]   (dense skinny GEMM, WMMA f32)
// block: 256 threads, output tile 64 x 128, grid.x = N/64
// ---------------------------------------------------------------------------
__global__ __launch_bounds__(256) void k_dense(const float* __restrict__ A,
                                               const float* __restrict__ B,
                                               float* __restrict__ C, int K) {
  __shared__ float As[64 * 68];
  __shared__ float Bp[32 * 256];  // K-pair packed: [kpair][n*2+parity]
  const int tid = threadIdx.x;
  const int lane = tid & 31;
  const int wid = tid >> 5;
  const int rowBase = (wid & 3) * 16;
  const int colBase = (wid >> 2) * 64;
  const int rowI = blockIdx.x * 64;
  const int ccB = tid >> 1;
  const int parB = tid & 1;
  v8f c0 = {}, c1 = {}, c2 = {}, c3 = {};
  for (int kc = 0; kc < K; kc += 64) {
#pragma unroll
    for (int q = 0; q < 16; ++q) {
      int idx = tid + 256 * q;
      int rr = idx >> 6, cc = idx & 63;
      As[rr * 68 + cc] = A[(size_t)(rowI + rr) * K + kc + cc];
    }
#pragma unroll
    for (int q = 0; q < 32; ++q) {  // kpair = q; LDS write fully coalesced
      Bp[q * 256 + tid] = B[(size_t)(kc + 2 * q + parB) * OUT + ccB];
    }
    __syncthreads();
#pragma unroll
    for (int k4 = 0; k4 < 64; k4 += 4) {
      v2f a = ldsA(As, 68, rowBase, k4, lane);
      c0 = wmma4(a, ldsBP(Bp, k4, colBase + 0, lane), c0);
      c1 = wmma4(a, ldsBP(Bp, k4, colBase + 16, lane), c1);
      c2 = wmma4(a, ldsBP(Bp, k4, colBase + 32, lane), c2);
      c3 = wmma4(a, ldsBP(Bp, k4, colBase + 48, lane), c3);
    }
    __syncthreads();
  }
  const int nl = lane & 15;
  const int mo = (lane >> 4) * 8;
  v8f cs[4] = {c0, c1, c2, c3};
#pragma unroll
  for (int t = 0; t < 4; ++t) {
#pragma unroll
    for (int v = 0; v < 8; ++v) {
      int row = rowI + rowBase + mo + v;
      int col = colBase + 16 * t + nl;
      C[(size_t)row * OUT + col] = cs[t][v];
    }
  }
}

// ---------------------------------------------------------------------------
// Fused streaming pass over A1/A2/A3 (reads each A once):
//   temp[i,j] = w0*A1 + w1*A2 + w2*A3          (written to workspace)
//   Fi[i,:]  += Ai[i,j] * token_i[j,:]         (TOK=10, shfl_xor reduction)
// block: 256 threads = 16 rows x 16 column groups, grid.x = N/16
// ---------------------------------------------------------------------------
__global__ __launch_bounds__(256) void k_stream(
    const float* __restrict__ A1, const float* __restrict__ A2,
    const float* __restrict__ A3, const float* __restrict__ T1,
    const float* __restrict__ T2, const float* __restrict__ T3,
    const float* __restrict__ wb, float* __restrict__ temp,
    float* __restrict__ F1, float* __restrict__ F2, float* __restrict__ F3) {
  __shared__ float tokS[3][256][TOK];
  const int tid = threadIdx.x;
  const int c = tid & 15;
  const int r = tid >> 4;
  const int i = blockIdx.x * 16 + r;
  const float w0 = wb[0], w1 = wb[1], w2 = wb[2];
  float acc1[TOK], acc2[TOK], acc3[TOK];
#pragma unroll
  for (int u = 0; u < TOK; ++u) acc1[u] = acc2[u] = acc3[u] = 0.f;

  for (int ch = 0; ch < N; ch += 256) {
    int j = ch + tid;
#pragma unroll
    for (int u = 0; u < TOK; ++u) {
      tokS[0][tid][u] = T1[(size_t)j * TOK + u];
      tokS[1][tid][u] = T2[(size_t)j * TOK + u];
      tokS[2][tid][u] = T3[(size_t)j * TOK + u];
    }
    __syncthreads();
#pragma unroll 4
    for (int kk = 0; kk < 16; ++kk) {
      int jl = c + 16 * kk;
      size_t idx = (size_t)i * N + ch + jl;
      float a1 = A1[idx], a2 = A2[idx], a3 = A3[idx];
      temp[idx] = w0 * a1 + w1 * a2 + w2 * a3;
#pragma unroll
      for (int u = 0; u < TOK; ++u) {
        acc1[u] = fmaf(a1, tokS[0][jl][u], acc1[u]);
        acc2[u] = fmaf(a2, tokS[1][jl][u], acc2[u]);
        acc3[u] = fmaf(a3, tokS[2][jl][u], acc3[u]);
      }
    }
    __syncthreads();
  }
  // reduce across the 16 column-group lanes (stays inside wave32 halves)
#pragma unroll
  for (int m = 8; m >= 1; m >>= 1) {
#pragma unroll
    for (int u = 0; u < TOK; ++u) {
      acc1[u] += __shfl_xor(acc1[u], m, 32);
      acc2[u] += __shfl_xor(acc2[u], m, 32);
      acc3[u] += __shfl_xor(acc3[u], m, 32);
    }
  }
  if (c == 0) {
#pragma unroll
    for (int u = 0; u < TOK; ++u) {
      F1[(size_t)i * TOK + u] = acc1[u];
      F2[(size_t)i * TOK + u] = acc2[u];
      F3[(size_t)i * TOK + u] = acc3[u];
    }
  }
}

// ---------------------------------------------------------------------------
// Cout[64-row block I] = (temp + temp^T) @ X + bias   (adj built per-tile)
//   adjS[b][a] = temp[I+b][J+a] + temp[J+a][I+b]
// mode 0: Cout = acc + bias                 (U1)
// mode 1: Cout = 0.5*(prevU + acc + bias)   (final output)
// block: 256 threads, output tile 64 x 128, grid.x = N/64
// ---------------------------------------------------------------------------
__global__ __launch_bounds__(256) void k_spmm(const float* __restrict__ temp,
                                              const float* __restrict__ X,
                                              const float* __restrict__ bias,
                                              const float* __restrict__ prevU,
                                              float* __restrict__ Cout,
                                              int mode) {
  __shared__ float adjS[64 * 68];
  __shared__ float Xp[32 * 256];  // K-pair packed: [kpair][n*2+parity]
  const int tid = threadIdx.x;
  const int lane = tid & 31;
  const int wid = tid >> 5;
  const int rowBase = (wid & 3) * 16;
  const int colBase = (wid >> 2) * 64;
  const int rowI = blockIdx.x * 64;
  const int ccB = tid >> 1;
  const int parB = tid & 1;
  v8f c0 = {}, c1 = {}, c2 = {}, c3 = {};
  for (int jc = 0; jc < N; jc += 64) {
#pragma unroll
    for (int q = 0; q < 16; ++q) {
      int idx = tid + 256 * q;
      int b = idx >> 6, a = idx & 63;
      float rv = temp[(size_t)(rowI + b) * N + (jc + a)];
      float cv = temp[(size_t)(jc + a) * N + (rowI + b)];
      adjS[b * 68 + a] = rv + cv;
    }
#pragma unroll
    for (int q = 0; q < 32; ++q) {  // kpair = q; LDS write fully coalesced
      Xp[q * 256 + tid] = X[(size_t)(jc + 2 * q + parB) * OUT + ccB];
    }
    __syncthreads();
#pragma unroll
    for (int k4 = 0; k4 < 64; k4 += 4) {
      v2f a = ldsA(adjS, 68, rowBase, k4, lane);
      c0 = wmma4(a, ldsBP(Xp, k4, colBase + 0, lane), c0);
      c1 = wmma4(a, ldsBP(Xp, k4, colBase + 16, lane), c1);
      c2 = wmma4(a, ldsBP(Xp, k4, colBase + 32, lane), c2);
      c3 = wmma4(a, ldsBP(Xp, k4, colBase + 48, lane), c3);
    }
    __syncthreads();
  }
  const int nl = lane & 15;
  const int mo = (lane >> 4) * 8;
  v8f cs[4] = {c0, c1, c2, c3};
#pragma unroll
  for (int t = 0; t < 4; ++t) {
#pragma unroll
    for (int v = 0; v < 8; ++v) {
      int row = rowI + rowBase + mo + v;
      int col = colBase + 16 * t + nl;
      float val = cs[t][v] + bias[col];
      if (mode == 1) val = 0.5f * (prevU[(size_t)row * OUT + col] + val);
      Cout[(size_t)row * OUT + col] = val;
    }
  }
}

extern "C" void kernel_launch(void* const* d_in, const int* in_sizes, int n_in,
                              void* d_out, int out_size, void* d_ws,
                              size_t ws_size, hipStream_t stream) {
  (void)in_sizes;
  (void)n_in;
  (void)out_size;
  (void)ws_size;
  const float* feature = (const float*)d_in[0];
  const float* A1 = (const float*)d_in[1];
  const float* A2 = (const float*)d_in[2];
  const float* A3 = (const float*)d_in[3];
  const float* T1 = (const float*)d_in[4];
  const float* T2 = (const float*)d_in[5];
  const float* T3 = (const float*)d_in[6];
  const float* wb = (const float*)d_in[7];
  const float* W1 = (const float*)d_in[8];
  const float* b1 = (const float*)d_in[9];
  const float* W2 = (const float*)d_in[10];
  const float* b2 = (const float*)d_in[11];

  float* out = (float*)d_out;
  float* F1 = out + (size_t)N * OUT;
  float* F2 = F1 + (size_t)N * TOK;
  float* F3 = F2 + (size_t)N * TOK;

  float* temp = (float*)d_ws;        // N*N f32 (256 MB)
  float* X1 = temp + (size_t)N * N;  // N*OUT
  float* U1 = X1 + (size_t)N * OUT;  // N*OUT
  float* Y2 = U1 + (size_t)N * OUT;  // N*OUT

  // 1) X1 = feature @ W1
  k_dense<<<N / 64, 256, 0, stream>>>(feature, W1, X1, NFEAT);
  // 2) temp + token features in one streaming pass over A1/A2/A3
  k_stream<<<N / 16, 256, 0, stream>>>(A1, A2, A3, T1, T2, T3, wb, temp, F1,
                                       F2, F3);
  // 3) U1 = adj @ X1 + b1
  k_spmm<<<N / 64, 256, 0, stream>>>(temp, X1, b1, nullptr, U1, 0);
  // 4) Y2 = U1 @ W2
  k_dense<<<N / 64, 256, 0, stream>>>(U1, W2, Y2, OUT);
  // 5) out = 0.5*(U1 + adj @ Y2 + b2)
  k_spmm<<<N / 64, 256, 0, stream>>>(temp, Y2, b2, U1, out, 1);
}